// N2PAttention_63333587747380
// MI455X (gfx1250) — compile-verified
//
#include <hip/hip_runtime.h>

typedef float v2f __attribute__((ext_vector_type(2)));
typedef float v4f __attribute__((ext_vector_type(4)));
typedef float v8f __attribute__((ext_vector_type(8)));

// ---- problem dims (fixed by the reference) ----
constexpr int Bb   = 2;
constexpr int C    = 128;
constexpr int T    = 32;
constexpr int Nn   = 512;
constexpr int Hh   = 4;
constexpr int Kk   = 16;
constexpr int HID  = 512;
constexpr int OUTC = 128;
constexpr int R    = T * Nn;     // 16384 (t,n) per batch
constexpr int BTN  = Bb * R;     // 32768 query columns
constexpr float EPSV   = 1e-5f;
constexpr float QSCALE = 0.17677669529663689f; // 1/sqrt(D=32)

constexpr int WSTR = 132; // LDS row stride (conflict-free + 8B aligned b64 frags)

__device__ __forceinline__ v2f ld2(const float* p) { return *(const v2f*)p; }

// D = A(16x4 f32) * B(4x16 f32) + C(16x16 f32), exact fp32 via WMMA
__device__ __forceinline__ v8f wmma4(v2f a, v2f b, v8f c) {
  return __builtin_amdgcn_wmma_f32_16x16x4_f32(false, a, false, b, (short)0, c,
                                               false, false);
}

// ---------------- kernel 0: x (B,C,T,N) -> xt (B*T*N, C) ----------------
__global__ void k_transpose(const float* __restrict__ x, float* __restrict__ xt) {
  int q = blockIdx.x;              // 0..BTN-1 == (b*T+t)*N+n
  int c = threadIdx.x;             // 0..127
  int b = q >> 14;
  int r = q & (R - 1);
  xt[q * C + c] = x[(b * C + c) * R + r];
}

// ---------------- kernel A: top-K=16 neighbor indices per query ----------------
__global__ void k_topk(const float* __restrict__ xt, int* __restrict__ idxb) {
  int wave = threadIdx.x >> 5;
  int lane = threadIdx.x & 31;
  int q = blockIdx.x * 8 + wave;   // one wave per query
  int b = q >> 14;
  int r = q & (R - 1);
  int t = r >> 9;

  const float* xq = xt + q * C;
  float a0 = xq[0], a1 = xq[1], a2 = xq[2];

  float dl[48];                    // 48*32 = 1536 pool columns
#pragma unroll
  for (int j = 0; j < 48; ++j) {
    int m  = j * 32 + lane;
    int s  = m >> 9;
    int nn = m & 511;
    int tt = t + s - 1; tt = tt < 0 ? 0 : (tt > T - 1 ? T - 1 : tt);
    const float* pc = xt + ((b * T + tt) * Nn + nn) * C;
    float d0 = a0 - pc[0], d1 = a1 - pc[1], d2 = a2 - pc[2];
    dl[j] = d0 * d0 + d1 * d1 + d2 * d2;
  }

  for (int kidx = 0; kidx < Kk; ++kidx) {
    float bd = dl[0];
    int bj = 0;
#pragma unroll
    for (int j = 1; j < 48; ++j) {   // within-lane: '<' keeps smaller j => smaller m
      if (dl[j] < bd) { bd = dl[j]; bj = j; }
    }
    int bm = bj * 32 + lane;
#pragma unroll
    for (int off = 16; off >= 1; off >>= 1) {   // cross-lane argmin, ties -> lower idx
      float od = __shfl_xor(bd, off, 32);
      int   om = __shfl_xor(bm, off, 32);
      if (od < bd || (od == bd && om < bm)) { bd = od; bm = om; }
    }
    if (lane == 0) idxb[q * Kk + kidx] = bm;
    if ((bm & 31) == lane) {
      int jj = bm >> 5;
#pragma unroll
      for (int j = 0; j < 48; ++j)
        if (j == jj) dl[j] = 3.402823466e+38f;
    }
  }
}

// ---------------- kernel B: attention + residual + BN -> h (BTN, C) ----------------
__global__ void __launch_bounds__(256) k_attn(
    const float* __restrict__ xt, const int* __restrict__ idxb,
    const float* __restrict__ Wq, const float* __restrict__ Wk,
    const float* __restrict__ Wv, const float* __restrict__ gamma,
    const float* __restrict__ beta, const float* __restrict__ mean,
    const float* __restrict__ var, float* __restrict__ hbuf) {
  extern __shared__ float sm[];
  float* Wq_l = sm;                       // 128 x 132
  float* Wk_l = Wq_l + C * WSTR;
  float* Wv_l = Wk_l + C * WSTR;
  float* wavebase = Wv_l + C * WSTR;      // 8 waves x (x|q|h columns)

  int tid  = threadIdx.x;
  int wave = tid >> 5, lane = tid & 31;
  int lrow = lane & 15, hi = lane >> 4;
  float* xcol = wavebase + wave * 384;
  float* qcol = xcol + 128;
  float* hcol = qcol + 128;

  for (int i = tid; i < C * C; i += 256) {  // weights -> LDS (padded rows)
    int rr = i >> 7, cc = i & 127;
    Wq_l[rr * WSTR + cc] = Wq[i];
    Wk_l[rr * WSTR + cc] = Wk[i];
    Wv_l[rr * WSTR + cc] = Wv[i];
  }
  float bsc[4], bbi[4];                     // BN constants for c = 4*lane+i
#pragma unroll
  for (int i = 0; i < 4; ++i) {
    int c = 4 * lane + i;
    float s = gamma[c] * rsqrtf(var[c] + EPSV);
    bsc[i] = s;
    bbi[i] = beta[c] - mean[c] * s;
  }
  __syncthreads();

  for (int it = 0; it < 8; ++it) {          // 8 queries per wave, uniform
    int q = blockIdx.x * 64 + wave * 8 + it;
    int b = q >> 14;
    int r = q & (R - 1);
    int t = r >> 9;

    *(v4f*)(xcol + 4 * lane) = *(const v4f*)(xt + q * C + 4 * lane);
    __syncthreads();

    // this lane's neighbor column (k = lane%16)
    int p  = idxb[q * Kk + lrow];
    int s  = p >> 9;
    int nn = p & 511;
    int tt = t + s - 1; tt = tt < 0 ? 0 : (tt > T - 1 ? T - 1 : tt);
    const float* nb = xt + ((b * T + tt) * Nn + nn) * C;

    // rel = x - nb, laid out as WMMA B fragments (32 K-chunks of 4)
    v2f bfrag[32];
#pragma unroll
    for (int ch = 0; ch < 32; ++ch) {
      int c0 = 4 * ch + 2 * hi;
      v2f xc = ld2(xcol + c0);
      v2f nv = ld2(nb + c0);
      bfrag[ch] = xc - nv;
    }

    // q = Wq * x (VALU; ~3% of FLOPs), pre-scaled by 1/sqrt(D)
    {
      float q0 = 0.f, q1 = 0.f, q2 = 0.f, q3 = 0.f;
      for (int c = 0; c < C; ++c) {
        float xv = xcol[c];
        q0 += Wq_l[(lane +  0) * WSTR + c] * xv;
        q1 += Wq_l[(lane + 32) * WSTR + c] * xv;
        q2 += Wq_l[(lane + 64) * WSTR + c] * xv;
        q3 += Wq_l[(lane + 96) * WSTR + c] * xv;
      }
      qcol[lane +  0] = q0 * QSCALE;
      qcol[lane + 32] = q1 * QSCALE;
      qcol[lane + 64] = q2 * QSCALE;
      qcol[lane + 96] = q3 * QSCALE;
    }
    __syncthreads();

#pragma unroll 1
    for (int h = 0; h < Hh; ++h) {          // head = 2 m-tiles of 16 rows
      v8f accV[2];
      float epart = 0.f;
#pragma unroll
      for (int half = 0; half < 2; ++half) {
        int m0 = 16 * (2 * h + half);
        v8f accK = {0, 0, 0, 0, 0, 0, 0, 0};
        v8f aV   = {0, 0, 0, 0, 0, 0, 0, 0};
#pragma unroll
        for (int ch = 0; ch < 32; ++ch) {   // K = Wk*rel, V = Wv*rel (fp32 WMMA)
          int c0 = 4 * ch + 2 * hi;
          v2f ak = ld2(Wk_l + (m0 + lrow) * WSTR + c0);
          v2f av = ld2(Wv_l + (m0 + lrow) * WSTR + c0);
          accK = wmma4(ak, bfrag[ch], accK);
          aV   = wmma4(av, bfrag[ch], aV);
        }
        accV[half] = aV;
#pragma unroll
        for (int j = 0; j < 8; ++j)         // energy partial (C-layout rows)
          epart += qcol[m0 + j + 8 * hi] * accK[j];
      }
      // energy[k] for k = lane%16, duplicated in both lane halves
      float e = epart + __shfl_xor(epart, 16, 32);
      float emax = e;
#pragma unroll
      for (int off = 8; off >= 1; off >>= 1)
        emax = fmaxf(emax, __shfl_xor(emax, off, 32));
      float pexp = __expf(e - emax);
      float psum = pexp;
#pragma unroll
      for (int off = 8; off >= 1; off >>= 1)
        psum += __shfl_xor(psum, off, 32);
      float attn = pexp / psum;

#pragma unroll
      for (int half = 0; half < 2; ++half) {  // agg = sum_k attn[k] * V[:,k]
        int m0 = 16 * (2 * h + half);
#pragma unroll
        for (int j = 0; j < 8; ++j) {
          float w = accV[half][j] * attn;
#pragma unroll
          for (int off = 8; off >= 1; off >>= 1)
            w += __shfl_xor(w, off, 32);
          if (lrow == 0) hcol[m0 + j + 8 * hi] = w;  // lanes 0 and 16 write
        }
      }
    }
    __syncthreads();

    {  // h = BN(x + agg) -> hbuf (BTN, C)
      v4f hv = *(v4f*)(hcol + 4 * lane);
      v4f xv = *(v4f*)(xcol + 4 * lane);
      v4f o;
#pragma unroll
      for (int i = 0; i < 4; ++i) o[i] = (hv[i] + xv[i]) * bsc[i] + bbi[i];
      *(v4f*)(hbuf + q * C + 4 * lane) = o;
    }
    __syncthreads();
  }
}

// ---------------- kernel C: y = W2 * leaky(W1 * h), fused per 64 columns ----------------
__global__ void __launch_bounds__(256) k_mlp(
    const float* __restrict__ hbuf, const float* __restrict__ W1,
    const float* __restrict__ W2, float* __restrict__ out) {
  extern __shared__ float sm[];
  float* hl  = sm;              // 64 cols x 132 (C padded)
  float* a1l = sm + 64 * 132;   // 64 cols x 516 (HID padded)

  int tid  = threadIdx.x;
  int wave = tid >> 5, lane = tid & 31;
  int lrow = lane & 15, hi = lane >> 4;
  int mbase = blockIdx.x * 64;

  for (int i = tid * 4; i < 64 * C; i += 1024) {  // h tile -> LDS
    int col = i >> 7, c = i & 127;
    *(v4f*)(hl + col * 132 + c) = *(const v4f*)(hbuf + (mbase + col) * C + c);
  }
  __syncthreads();

  // stage 1: a1(HID x 64) = leaky_relu(W1 * h)
  for (int job = wave; job < 128; job += 8) {   // 32 m-tiles x 4 col-tiles
    int mt = job >> 2, ct = job & 3;
    int o0 = 16 * mt, colb = 16 * ct;
    v8f acc = {0, 0, 0, 0, 0, 0, 0, 0};
#pragma unroll
    for (int ch = 0; ch < 32; ++ch) {
      int c0 = 4 * ch + 2 * hi;
      v2f a = ld2(W1 + (o0 + lrow) * C + c0);
      v2f bb = ld2(hl + (colb + lrow) * 132 + c0);
      acc = wmma4(a, bb, acc);
    }
    int col = colb + lrow;
#pragma unroll
    for (int j = 0; j < 8; ++j) {
      float v = acc[j];
      v = v > 0.f ? v : 0.2f * v;               // leaky ReLU 0.2
      a1l[col * 516 + o0 + j + 8 * hi] = v;
    }
  }
  __syncthreads();

  // stage 2: y(128 x 64) = W2 * a1 -> out channels 4..131
  int b   = mbase >> 14;
  int rr0 = mbase & (R - 1);
  for (int job = wave; job < 32; job += 8) {    // 8 m-tiles x 4 col-tiles
    int mt = job >> 2, ct = job & 3;
    int o0 = 16 * mt, colb = 16 * ct;
    v8f acc = {0, 0, 0, 0, 0, 0, 0, 0};
#pragma unroll 4
    for (int ch = 0; ch < 128; ++ch) {
      int c0 = 4 * ch + 2 * hi;
      v2f a = ld2(W2 + (o0 + lrow) * HID + c0);
      v2f bb = ld2(a1l + (colb + lrow) * 516 + c0);
      acc = wmma4(a, bb, acc);
    }
    int rr = rr0 + colb + lrow;
#pragma unroll
    for (int j = 0; j < 8; ++j) {
      int o = o0 + j + 8 * hi;
      out[b * (132 * R) + (4 + o) * R + rr] = acc[j];
    }
  }
}

// ---------------- kernel D: out[:, :4] = x[:, :4] ----------------
__global__ void k_copy4(const float* __restrict__ x, float* __restrict__ out) {
  int i  = blockIdx.x * 256 + threadIdx.x;  // 0 .. 2*4*16384-1
  int b  = i >> 16;
  int rr = i & 65535;
  int ch = rr >> 14;
  int r  = rr & (R - 1);
  out[b * (132 * R) + ch * R + r] = x[(b * C + ch) * R + r];
}

extern "C" void kernel_launch(void* const* d_in, const int* in_sizes, int n_in,
                              void* d_out, int out_size, void* d_ws, size_t ws_size,
                              hipStream_t stream) {
  (void)in_sizes; (void)n_in; (void)out_size; (void)ws_size;
  const float* x  = (const float*)d_in[0];
  const float* Wq = (const float*)d_in[1];
  const float* Wk = (const float*)d_in[2];
  const float* Wv = (const float*)d_in[3];
  const float* W1 = (const float*)d_in[4];
  const float* W2 = (const float*)d_in[5];
  const float* gg = (const float*)d_in[6];
  const float* bt = (const float*)d_in[7];
  const float* mn = (const float*)d_in[8];
  const float* vr = (const float*)d_in[9];
  float* out = (float*)d_out;

  // workspace: xt (16.8MB) | idx (2MB) | h (16.8MB)  => ~35.7MB
  char*  ws   = (char*)d_ws;
  float* xt   = (float*)ws;
  int*   idxb = (int*)(ws + (size_t)BTN * C * 4);
  float* hbuf = (float*)(ws + (size_t)BTN * C * 4 + (size_t)BTN * Kk * 4);

  k_transpose<<<BTN, C, 0, stream>>>(x, xt);
  k_topk<<<BTN / 8, 256, 0, stream>>>(xt, idxb);

  size_t shB = (size_t)(3 * C * WSTR + 8 * 384) * sizeof(float);  // 215,040 B
  k_attn<<<BTN / 64, 256, shB, stream>>>(xt, idxb, Wq, Wk, Wv, gg, bt, mn, vr, hbuf);

  size_t shC = (size_t)(64 * 132 + 64 * 516) * sizeof(float);     // 165,888 B
  k_mlp<<<BTN / 64, 256, shC, stream>>>(hbuf, W1, W2, out);

  k_copy4<<<(Bb * 4 * R) / 256, 256, 0, stream>>>(x, out);
}